// VectorQuantizer_87703232184514
// MI455X (gfx1250) — compile-verified
//
#include <hip/hip_runtime.h>

typedef float v2f __attribute__((ext_vector_type(2)));
typedef float v4f __attribute__((ext_vector_type(4)));
typedef float v8f __attribute__((ext_vector_type(8)));

#define N_ROWS 65536
#define DDIM 256
#define KCODES 1024
#define ROWS_PER_WG 128
#define ROWS_PER_WAVE 16
#define TILE_CODES 16
#define NTILES (KCODES / TILE_CODES)   // 64
#define KSTEPS (DDIM / 4)              // 64 wmma steps per tile
#define LDSB_STRIDE 260                // 256 + 4 pad floats (bank-conflict-free b64 frags)

// ---------------------------------------------------------------------------
// Kernel 0: e2[k] = ||codebook[k]||^2 ; zero counts + commit accumulator.
// Must run every call (graph replay does not re-zero workspace).
// ---------------------------------------------------------------------------
__global__ void vq_init(const float* __restrict__ cb, float* __restrict__ ws) {
  int k = blockIdx.x * blockDim.x + threadIdx.x;
  if (k < KCODES) {
    const v4f* row = (const v4f*)(cb + (size_t)k * DDIM);
    float s = 0.f;
#pragma unroll 8
    for (int j = 0; j < DDIM / 4; ++j) {
      v4f v = row[j];
      s += v.x * v.x + v.y * v.y + v.z * v.z + v.w * v.w;
    }
    ws[k] = s;             // e2
    ws[KCODES + k] = 0.f;  // counts
  }
  if (blockIdx.x == 0 && threadIdx.x == 0) ws[2 * KCODES] = 0.f;  // commit sum
}

// ---------------------------------------------------------------------------
// Kernel 1: main — WMMA GEMM tiles + argmin + gather + stats.
// ---------------------------------------------------------------------------
__launch_bounds__(256)
__global__ void vq_main(const float* __restrict__ x, const float* __restrict__ cb,
                        const float* __restrict__ e2c, float* __restrict__ counts,
                        float* __restrict__ commit, float* __restrict__ y,
                        float* __restrict__ oidx) {
  __shared__ float ldsB[2 * TILE_CODES * LDSB_STRIDE];
  __shared__ float ldsE[KCODES];

  const int tid  = threadIdx.x;
  const int w    = tid >> 5;       // wave id within workgroup (wave32)
  const int lane = tid & 31;
  const int n    = lane & 15;      // column within 16-wide tile (B/C layout)
  const int h    = lane >> 4;      // half-wave (K split for A/B fragments)

  const int rowbase = blockIdx.x * ROWS_PER_WG + w * ROWS_PER_WAVE;
  const int arow    = rowbase + n;  // A-fragment row owned by this lane

  // Stage e2 into LDS (covered by first __syncthreads in the tile loop).
  for (int i = tid; i < KCODES; i += 256) ldsE[i] = e2c[i];

  // A fragments in registers: a[s] holds x[arow][4s + 2h + {0,1}]  (16x4 f32 layout)
  v2f a[KSTEPS];
  const float* xr = x + (size_t)arow * DDIM + 2 * h;
  float x2p = 0.f;
#pragma unroll
  for (int s = 0; s < KSTEPS; ++s) {
    a[s] = *(const v2f*)(xr + 4 * s);
    x2p += a[s].x * a[s].x + a[s].y * a[s].y;
  }
  // ||x_row||^2: combine the two half-wave partials; lane m then holds row m's sum.
  float x2 = x2p + __shfl_xor(x2p, 16, 32);

  // Preload codebook tile 0 into LDS buffer 0 (cooperative, coalesced b128).
  {
    const int c = tid >> 4, seg = tid & 15;
    const v4f* src = (const v4f*)(cb + (size_t)c * DDIM + seg * 16);
    v4f* dst = (v4f*)(ldsB + c * LDSB_STRIDE + seg * 16);
#pragma unroll
    for (int j = 0; j < 4; ++j) dst[j] = src[j];
  }

  float best[8];
  int bidx[8];
#pragma unroll
  for (int r = 0; r < 8; ++r) { best[r] = 3.4e38f; bidx[r] = 0; }

  for (int t = 0; t < NTILES; ++t) {
    __syncthreads();  // buf[t&1] fully written; previous reads of buf[(t&1)^1] done
    const int buf = t & 1;
    const int c = tid >> 4, seg = tid & 15;

    // Prefetch next tile to registers (overlaps with WMMA work).
    v4f pf[4];
    if (t + 1 < NTILES) {
      const v4f* src = (const v4f*)(cb + (size_t)((t + 1) * TILE_CODES + c) * DDIM + seg * 16);
#pragma unroll
      for (int j = 0; j < 4; ++j) pf[j] = src[j];
    }

    // sim tile: 64 x v_wmma_f32_16x16x4_f32, split into TWO independent
    // accumulation chains (even/odd k-steps) so consecutive WMMAs never
    // serialize on the same accumulator registers.
    v8f acc0 = {0.f, 0.f, 0.f, 0.f, 0.f, 0.f, 0.f, 0.f};
    v8f acc1 = {0.f, 0.f, 0.f, 0.f, 0.f, 0.f, 0.f, 0.f};
    const float* bp = ldsB + buf * (TILE_CODES * LDSB_STRIDE) + n * LDSB_STRIDE + 2 * h;
#pragma unroll
    for (int s = 0; s < KSTEPS; s += 2) {
      v2f b0 = *(const v2f*)(bp + 4 * s);        // ds_load_2addr_b64 pairs
      v2f b1 = *(const v2f*)(bp + 4 * (s + 1));
      acc0 = __builtin_amdgcn_wmma_f32_16x16x4_f32(
          false, a[s], false, b0, (short)0, acc0, false, false);
      acc1 = __builtin_amdgcn_wmma_f32_16x16x4_f32(
          false, a[s + 1], false, b1, (short)0, acc1, false, false);
    }

    // Drain prefetch into the other LDS buffer.
    if (t + 1 < NTILES) {
      v4f* dst = (v4f*)(ldsB + (buf ^ 1) * (TILE_CODES * LDSB_STRIDE) + c * LDSB_STRIDE + seg * 16);
#pragma unroll
      for (int j = 0; j < 4; ++j) dst[j] = pf[j];
    }

    // dist(partial) = e2 - 2*sim ; running argmin per (row, lane-column).
    const float e2v = ldsE[t * TILE_CODES + n];
    const int code = t * TILE_CODES + n;
#pragma unroll
    for (int r = 0; r < 8; ++r) {
      float sim = acc0[r] + acc1[r];
      float d = __builtin_fmaf(-2.f, sim, e2v);
      if (d < best[r]) { best[r] = d; bidx[r] = code; }
    }
  }

  // Butterfly min-reduce across the 16 lane-columns (ties -> lower index, like argmin).
#pragma unroll
  for (int m = 8; m >= 1; m >>= 1) {
#pragma unroll
    for (int r = 0; r < 8; ++r) {
      float od = __shfl_xor(best[r], m, 32);
      int oi = __shfl_xor(bidx[r], m, 32);
      if (od < best[r] || (od == best[r] && oi < bidx[r])) { best[r] = od; bidx[r] = oi; }
    }
  }
  // Now lanes 0-15 all hold rows 0..7 results (reg r), lanes 16-31 rows 8..15.

  float csum = 0.f;
#pragma unroll
  for (int rr = 0; rr < 16; ++rr) {
    const int srcl = (rr < 8) ? 0 : 16;
    const int code = __shfl(bidx[rr & 7], srcl, 32);
    const float bd = __shfl(best[rr & 7], srcl, 32);
    const float x2r = __shfl(x2, rr, 32);
    csum += bd + x2r;  // full squared distance of row rr = commit contribution

    const int row = rowbase + rr;
    // y[row] = codebook[code]  (coalesced 2x b128 per lane)
    const v4f* q = (const v4f*)(cb + (size_t)code * DDIM + lane * 8);
    v4f* yo = (v4f*)(y + (size_t)row * DDIM + lane * 8);
    yo[0] = q[0];
    yo[1] = q[1];
    if (lane == 0) {
      oidx[row] = (float)code;
      atomicAdd(&counts[code], 1.0f);  // exact: integer-valued f32 < 2^24
    }
  }
  if (lane == 0) atomicAdd(commit, csum);
}

// ---------------------------------------------------------------------------
// Kernel 2: entropy / perplexity / usage / loss from counts + commit sum.
// ---------------------------------------------------------------------------
__global__ void vq_fin(const float* __restrict__ ws, float* __restrict__ sc) {
  __shared__ float rh[1024];
  __shared__ float ru[1024];
  const int k = threadIdx.x;
  const float c = ws[KCODES + k];
  const float p = c * (1.0f / (float)N_ROWS);
  rh[k] = -p * log2f(p + 1e-10f);
  ru[k] = (c > 0.f) ? 1.f : 0.f;
  __syncthreads();
  for (int s = 512; s >= 1; s >>= 1) {
    if (k < s) { rh[k] += rh[k + s]; ru[k] += ru[k + s]; }
    __syncthreads();
  }
  if (k == 0) {
    const float H = rh[0];
    const float mse = ws[2 * KCODES] * (1.0f / ((float)N_ROWS * (float)DDIM));
    sc[0] = 1.25f * mse;            // loss = BETA*commit + codebook (numerically equal)
    sc[1] = exp2f(H);               // perplexity = exp(H*ln2)
    sc[2] = ru[0] / (float)KCODES;  // usage
    sc[3] = H;
  }
}

// ---------------------------------------------------------------------------
extern "C" void kernel_launch(void* const* d_in, const int* in_sizes, int n_in,
                              void* d_out, int out_size, void* d_ws, size_t ws_size,
                              hipStream_t stream) {
  (void)in_sizes; (void)n_in; (void)out_size; (void)ws_size;
  const float* x = (const float*)d_in[0];
  const float* cb = (const float*)d_in[1];
  float* out = (float*)d_out;
  float* ws = (float*)d_ws;  // [0..1023] e2 | [1024..2047] counts | [2048] commit

  float* y = out;                               // [N_ROWS * DDIM]
  float* oidx = out + (size_t)N_ROWS * DDIM;    // [N_ROWS] (as float)
  float* scalars = oidx + N_ROWS;               // loss, perplexity, usage, H

  vq_init<<<(KCODES + 255) / 256, 256, 0, stream>>>(cb, ws);
  vq_main<<<N_ROWS / ROWS_PER_WG, 256, 0, stream>>>(
      x, cb, ws, ws + KCODES, ws + 2 * KCODES, y, oidx);
  vq_fin<<<1, 1024, 0, stream>>>(ws, scalars);
}